// KoopmanLQR_87024627352059
// MI455X (gfx1250) — compile-verified
//
#include <hip/hip_runtime.h>
#include <cmath>

typedef float v2f __attribute__((ext_vector_type(2)));
typedef float v8f __attribute__((ext_vector_type(8)));

// D = A(16x4) * B(4x16) + C, all f32, wave32.  Layout per CDNA5 ISA 7.12.2:
//  A: lanes 0-15 -> M=lane, VGPR0=K(0),VGPR1=K(1); lanes 16-31 -> M=lane-16, K(2),K(3)
//  B: lanes 0-15 -> N=lane, VGPR0=K0,VGPR1=K1;     lanes 16-31 -> N=lane-16, K2,K3
//  C/D: VGPR r -> M = r (lanes 0-15) / r+8 (lanes 16-31), N = lane&15
__device__ __forceinline__ v8f wmma_f32(v2f a, v2f b, v8f c) {
  return __builtin_amdgcn_wmma_f32_16x16x4_f32(false, a, false, b, (short)0, c,
                                               false, false);
}

// ---------------------------------------------------------------------------
// prep: Qd = exp(q), Rd = exp(r), gg = W_phi@x_goal + b_phi, V = diag(Qd),
//       v = Qd*gg, U = [A | B] (256 x 320)
// ---------------------------------------------------------------------------
__global__ void prep_kernel(const float* __restrict__ Wphi, const float* __restrict__ bphi,
                            const float* __restrict__ xgoal,
                            const float* __restrict__ Amat, const float* __restrict__ Bmat,
                            const float* __restrict__ qlog, const float* __restrict__ rlog,
                            float* __restrict__ V, float* __restrict__ U,
                            float* __restrict__ vv, float* __restrict__ Qd,
                            float* __restrict__ Rd, float* __restrict__ gg)
{
  __shared__ float sQ[256];
  const int t = threadIdx.x;  // 320 threads
  if (t < 256) {
    float q = expf(qlog[t]);
    sQ[t] = q;
    Qd[t] = q;
    float s = bphi[t];
    for (int c = 0; c < 128; ++c) s += Wphi[t * 128 + c] * xgoal[c];
    gg[t] = s;
    vv[t] = q * s;
  } else {
    Rd[t - 256] = expf(rlog[t - 256]);
  }
  __syncthreads();
  for (int idx = t; idx < 256 * 256; idx += 320) {
    int i = idx >> 8, j = idx & 255;
    V[idx] = (i == j) ? sQ[i] : 0.0f;
  }
  for (int idx = t; idx < 256 * 320; idx += 320) {
    int r = idx / 320, c = idx - r * 320;
    U[idx] = (c < 256) ? Amat[r * 256 + c] : Bmat[r * 64 + (c - 256)];
  }
}

// ---------------------------------------------------------------------------
// Generic f32 WMMA GEMM, one 16x16 tile per wave32 block.
// TA=false: O = X @ Y ;  TA=true: O = X^T @ Y   (Kdim multiple of 4)
// ---------------------------------------------------------------------------
template <bool TA>
__global__ void gemm_wmma(const float* __restrict__ X, const float* __restrict__ Y,
                          float* __restrict__ O, int ldx, int ldy, int ldo, int Kdim)
{
  const int lane = threadIdx.x;
  const int lr = lane & 15, half = lane >> 4, koff = half * 2;
  const int m0 = blockIdx.y * 16, n0 = blockIdx.x * 16;
  v8f acc = {};
  for (int k = 0; k < Kdim; k += 4) {
    v2f a, b;
    if (TA) {
      a.x = X[(k + koff) * ldx + m0 + lr];
      a.y = X[(k + koff + 1) * ldx + m0 + lr];
    } else {
      const float* xr = X + (m0 + lr) * ldx + k + koff;
      a.x = xr[0];
      a.y = xr[1];
    }
    b.x = Y[(k + koff) * ldy + n0 + lr];
    b.y = Y[(k + koff + 1) * ldy + n0 + lr];
    acc = wmma_f32(a, b, acc);
  }
#pragma unroll
  for (int r = 0; r < 8; ++r)
    O[(m0 + r + half * 8) * ldo + n0 + lr] = acc[r];
}

// ---------------------------------------------------------------------------
// stage34 (single block, 256 threads):
//   av = A^T v, bv = B^T v
//   M = O2[256+i][256+j] + diag(Rd) ; Minv via in-LDS Gauss-Jordan (SPD)
//   K = Minv @ N   (N = O2[256+j][0:256]) ; kv = Minv @ bv
// ---------------------------------------------------------------------------
__global__ void stage34_kernel(const float* __restrict__ Amat, const float* __restrict__ Bmat,
                               const float* __restrict__ vv, const float* __restrict__ O2,
                               const float* __restrict__ Rd,
                               float* __restrict__ Kmat, float* __restrict__ kv,
                               float* __restrict__ av, float* __restrict__ bv)
{
  __shared__ float aug[64][130];  // [M | I] -> [I | Minv]
  __shared__ float sbv[64];
  const int t = threadIdx.x;  // 256

  {  // av = A^T v (256), bv = B^T v (64)
    float s = 0.0f;
    for (int r = 0; r < 256; ++r) s += Amat[r * 256 + t] * vv[r];
    av[t] = s;
  }
  if (t < 64) {
    float s = 0.0f;
    for (int r = 0; r < 256; ++r) s += Bmat[r * 64 + t] * vv[r];
    bv[t] = s;
    sbv[t] = s;
  }
  for (int idx = t; idx < 64 * 64; idx += 256) {
    int i = idx >> 6, j = idx & 63;
    aug[i][j] = O2[(256 + i) * 320 + 256 + j] + ((i == j) ? Rd[i] : 0.0f);
    aug[i][64 + j] = (i == j) ? 1.0f : 0.0f;
  }
  __syncthreads();

  // Gauss-Jordan, no pivoting (M is SPD)
  const int row = t >> 2;       // 4 threads per row
  const int c4 = t & 3;
  for (int p = 0; p < 64; ++p) {
    float inv = 1.0f / aug[p][p];
    __syncthreads();
    if (t < 128) aug[p][t] *= inv;
    __syncthreads();
    float f = aug[row][p];
    __syncthreads();
    if (row != p) {
      for (int j = c4; j < 128; j += 4) aug[row][j] -= f * aug[p][j];
    }
    __syncthreads();
  }

  // K[j][n] = sum_q Minv[j][q] * O2[(256+q)*320 + n]
  for (int idx = t; idx < 64 * 256; idx += 256) {
    int j = idx >> 8, n = idx & 255;  // per pass: all threads share j (LDS broadcast)
    float s = 0.0f;
    for (int q = 0; q < 64; ++q) s += aug[j][64 + q] * O2[(256 + q) * 320 + n];
    Kmat[j * 256 + n] = s;
  }
  if (t < 64) {
    float s = 0.0f;
    for (int q = 0; q < 64; ++q) s += aug[t][64 + q] * sbv[q];
    kv[t] = s;
  }
}

// ---------------------------------------------------------------------------
// stage5 (grid 16x17, 32 threads):
//   by<16 : V' = G - P@K + diag(Qd)   (G=O2[:,0:256], P=O2[:,256:320], WMMA k=64)
//   by==16,bx==0 : v' = av - K^T bv + Qd*gg
// ---------------------------------------------------------------------------
__global__ void stage5_kernel(const float* __restrict__ O2, const float* __restrict__ Kmat,
                              const float* __restrict__ Qd, const float* __restrict__ gg,
                              const float* __restrict__ av, const float* __restrict__ bv,
                              float* __restrict__ V, float* __restrict__ vv)
{
  const int lane = threadIdx.x;
  if (blockIdx.y == 16) {
    if (blockIdx.x != 0) return;
    for (int i = lane; i < 256; i += 32) {
      float s = av[i] + Qd[i] * gg[i];
      for (int j = 0; j < 64; ++j) s -= Kmat[j * 256 + i] * bv[j];
      vv[i] = s;
    }
    return;
  }
  const int lr = lane & 15, half = lane >> 4, koff = half * 2;
  const int m0 = blockIdx.y * 16, n0 = blockIdx.x * 16;
  v8f acc = {};
  for (int k = 0; k < 64; k += 4) {
    v2f a, b;
    const float* pr = O2 + (m0 + lr) * 320 + 256 + k + koff;
    a.x = pr[0];
    a.y = pr[1];
    b.x = Kmat[(k + koff) * 256 + n0 + lr];
    b.y = Kmat[(k + koff + 1) * 256 + n0 + lr];
    acc = wmma_f32(a, b, acc);
  }
#pragma unroll
  for (int r = 0; r < 8; ++r) {
    int m = m0 + r + half * 8, n = n0 + lr;
    float val = O2[m * 320 + n] - acc[r];
    if (m == n) val += Qd[m];
    V[m * 256 + n] = val;
  }
}

// ---------------------------------------------------------------------------
// finalize: C = K0 @ W_phi (64x128), d = kv - K0 @ b_phi (64)
// ---------------------------------------------------------------------------
__global__ void finalize_kernel(const float* __restrict__ Wphi, const float* __restrict__ bphi,
                                const float* __restrict__ Kmat, const float* __restrict__ kv,
                                float* __restrict__ Cmat, float* __restrict__ dvec)
{
  const int t = threadIdx.x;  // 256
  if (blockIdx.x < 32) {
    int j = blockIdx.x * 2 + (t >> 7);
    int c = t & 127;
    float s = 0.0f;
    for (int i = 0; i < 256; ++i) s += Kmat[j * 256 + i] * Wphi[i * 128 + c];
    Cmat[j * 128 + c] = s;
  } else if (t < 64) {
    float s = kv[t];
    for (int i = 0; i < 256; ++i) s -= Kmat[t * 256 + i] * bphi[i];
    dvec[t] = s;
  }
}

// ---------------------------------------------------------------------------
// batch: out = d - x0 @ C^T   (B x 64), memory-bound -> 96 MB @ 23.3 TB/s.
// 256 threads = 8 waves; wave handles 16 rows x 64 cols; C in padded LDS.
// ---------------------------------------------------------------------------
__global__ void batch_kernel(const float* __restrict__ x0, const float* __restrict__ Cmat,
                             const float* __restrict__ dvec, float* __restrict__ out)
{
  __shared__ float Cs[64 * 132];  // row n at stride 132 -> conflict-free B reads
  __shared__ float ds[64];
  const int t = threadIdx.x;
  for (int idx = t; idx < 64 * 128; idx += 256) {
    int n = idx >> 7, c = idx & 127;
    Cs[n * 132 + c] = Cmat[idx];
  }
  if (t < 64) ds[t] = dvec[t];
  __syncthreads();

  const int wave = t >> 5, lane = t & 31;
  const int lr = lane & 15, half = lane >> 4, koff = half * 2;
  const size_t m0 = ((size_t)blockIdx.x * 8 + wave) * 16;

  v8f acc[4] = {};
  const float* xrow = x0 + (m0 + lr) * 128;
#pragma unroll 4
  for (int k = 0; k < 128; k += 4) {
    v2f a;
    a.x = xrow[k + koff];
    a.y = xrow[k + koff + 1];
#pragma unroll
    for (int nt = 0; nt < 4; ++nt) {
      v2f b;
      b.x = Cs[(nt * 16 + lr) * 132 + k + koff];
      b.y = Cs[(nt * 16 + lr) * 132 + k + koff + 1];
      acc[nt] = wmma_f32(a, b, acc[nt]);
    }
  }
#pragma unroll
  for (int nt = 0; nt < 4; ++nt) {
    float dval = ds[nt * 16 + lr];
#pragma unroll
    for (int r = 0; r < 8; ++r) {
      size_t m = m0 + r + half * 8;
      out[m * 64 + nt * 16 + lr] = dval - acc[nt][r];
    }
  }
}

// ---------------------------------------------------------------------------
extern "C" void kernel_launch(void* const* d_in, const int* in_sizes, int n_in,
                              void* d_out, int out_size, void* d_ws, size_t ws_size,
                              hipStream_t stream)
{
  const float* x0    = (const float*)d_in[0];
  const float* Wphi  = (const float*)d_in[1];
  const float* bphi  = (const float*)d_in[2];
  const float* xgoal = (const float*)d_in[3];
  const float* Amat  = (const float*)d_in[4];  // (256,256)
  const float* Bmat  = (const float*)d_in[5];  // (256,64)
  const float* qlog  = (const float*)d_in[6];
  const float* rlog  = (const float*)d_in[7];

  float* ws = (float*)d_ws;  // ~1.4 MB of f32 scratch
  float* V  = ws;             // 256*256
  float* U  = V + 256 * 256;  // 256*320  [A|B]
  float* Z  = U + 256 * 320;  // 256*320  V@U
  float* O2 = Z + 256 * 320;  // 320*320  U^T@Z = [[G,P],[N,M-R]]
  float* Km = O2 + 320 * 320; // 64*256
  float* Cm = Km + 64 * 256;  // 64*128
  float* vv = Cm + 64 * 128;  // 256
  float* av = vv + 256;       // 256
  float* bv = av + 256;       // 64
  float* kv = bv + 64;        // 64
  float* Qd = kv + 64;        // 256
  float* Rd = Qd + 256;       // 64
  float* gg = Rd + 64;        // 256
  float* dv = gg + 256;       // 64

  prep_kernel<<<1, 320, 0, stream>>>(Wphi, bphi, xgoal, Amat, Bmat, qlog, rlog,
                                     V, U, vv, Qd, Rd, gg);

  for (int tstep = 0; tstep < 200; ++tstep) {
    // Z = V @ U           (256 x 320, k=256)
    gemm_wmma<false><<<dim3(20, 16), 32, 0, stream>>>(V, U, Z, 256, 320, 320, 256);
    // O2 = U^T @ Z        (320 x 320, k=256)
    gemm_wmma<true><<<dim3(20, 20), 32, 0, stream>>>(U, Z, O2, 320, 320, 320, 256);
    // invert M, K = Minv@N, kv, av, bv
    stage34_kernel<<<1, 256, 0, stream>>>(Amat, Bmat, vv, O2, Rd, Km, kv, av, bv);
    // V' = G - P@K + Q ; v' = av - K^T bv + Q*gg
    stage5_kernel<<<dim3(16, 17), 32, 0, stream>>>(O2, Km, Qd, gg, av, bv, V, vv);
  }

  finalize_kernel<<<33, 256, 0, stream>>>(Wphi, bphi, Km, kv, Cm, dv);

  const int Btot = in_sizes[0] / 128;           // 131072
  batch_kernel<<<Btot / 128, 256, 0, stream>>>(x0, Cm, dv, (float*)d_out);
}